// ScatterConnection_5841155523190
// MI455X (gfx1250) — compile-verified
//
#include <hip/hip_runtime.h>
#include <stdint.h>

// Problem constants fixed by setup_inputs():
//   x: (B=32, M=512, N=256) f32, location: (B, M, 2) i64 in [0,128)
//   out: (B, N, H, W) = (32, 256, 128, 128) f32  -> 512 MB
#define B_   32
#define M_   512
#define N_   256
#define Wdim 128
#define HW_  16384  // 128*128

// Native 16-byte vector type (HIP's float4 class is rejected by
// __builtin_nontemporal_store).
typedef float v4f __attribute__((ext_vector_type(4)));

// ---------------------------------------------------------------------------
// Kernel 1: zero the 512 MB output with wide non-temporal stores.
// Pure streaming fill: 23.3 TB/s HBM floor ~22 us. NT hint keeps the bulk
// fill from occupying L2 with lines that will never be re-read (only ~270 MB
// of lines are later RMW'd by atomics; L2 handles those on demand).
// ---------------------------------------------------------------------------
__global__ void ScatterConnection_zero_kernel(v4f* __restrict__ out, size_t n4) {
    size_t i      = (size_t)blockIdx.x * blockDim.x + threadIdx.x;
    size_t stride = (size_t)gridDim.x * blockDim.x;
    const v4f z = {0.f, 0.f, 0.f, 0.f};
    for (; i < n4; i += stride) {
        __builtin_nontemporal_store(z, &out[i]);   // global_store_b128 ... th:TH_STORE_NT
    }
}

// ---------------------------------------------------------------------------
// Kernel 2: one 256-thread block (8 wave32s) per entity (b, m).
//   - async-copy the entity's 256 floats (1 KB) global -> LDS
//     (CDNA5 GLOBAL_LOAD_ASYNC_TO_LDS_B128, tracked by ASYNCcnt)
//   - thread t = channel n does one hardware f32 atomic add into
//     out[b, n, p] at flat index ((b*N + n)*HW + p).
// Duplicate locations must accumulate -> atomics are required for
// correctness; lanes hit 64 KB-strided lines so atomics spread across L2.
// ---------------------------------------------------------------------------
__global__ void ScatterConnection_scatter_kernel(const float* __restrict__ x,
                                                 const long long* __restrict__ loc,
                                                 float* __restrict__ out) {
    __shared__ float tile[N_];
    __shared__ int   p_sh;

    const int e = blockIdx.x;   // entity index in [0, B*M)
    const int t = threadIdx.x;  // channel n in [0, N)

    if (t == 0) {
        // location is (y, x) int64; flat spatial index p = y*W + x
        long long y  = loc[2LL * e + 0];
        long long xx = loc[2LL * e + 1];
        p_sh = (int)(y * (long long)Wdim + xx);
    }

    // Waves 0 and 1 (threads 0..63, EXEC all-ones within each wave) issue
    // 64 x 16B async loads covering the 1 KB embedding, then drain ASYNCcnt.
    if (t < 64) {
        const float* src = x + (size_t)e * N_ + (size_t)t * 4;
        // Low 32 bits of a flat LDS pointer are the LDS byte offset
        // (aperture base lives in the high dword; ISA 10.2 truncates).
        unsigned lds_off = (unsigned)(uintptr_t)(&tile[0]) + (unsigned)t * 16u;
        asm volatile("global_load_async_to_lds_b128 %0, %1, off"
                     :
                     : "v"(lds_off), "v"(src)
                     : "memory");
        asm volatile("s_wait_asynccnt 0" ::: "memory");
    }
    __syncthreads();

    const int   p = p_sh;
    const float v = tile[t];

    const int b = e / M_;  // M_ = 512 -> shift
    size_t oidx = ((size_t)b * N_ + (size_t)t) * (size_t)HW_ + (size_t)p;
    unsafeAtomicAdd(&out[oidx], v);   // global_atomic_add_f32 (no CAS loop)
}

extern "C" void kernel_launch(void* const* d_in, const int* in_sizes, int n_in,
                              void* d_out, int out_size, void* d_ws, size_t ws_size,
                              hipStream_t stream) {
    (void)in_sizes; (void)n_in; (void)d_ws; (void)ws_size;

    const float*     x   = (const float*)d_in[0];
    const long long* loc = (const long long*)d_in[1];  // int64 per reference
    float*           out = (float*)d_out;

    // 1) zero 512 MB output (out_size = B*N*H*W, divisible by 4)
    size_t n4 = (size_t)out_size / 4;
    ScatterConnection_zero_kernel<<<4096, 256, 0, stream>>>((v4f*)out, n4);

    // 2) scatter-add: one block per entity
    ScatterConnection_scatter_kernel<<<B_ * M_, 256, 0, stream>>>(x, loc, out);
}